// MPS_8186207666266
// MI455X (gfx1250) — compile-verified
//
#include <hip/hip_runtime.h>
#include <hip/hip_bf16.h>
#include <stdint.h>

typedef __attribute__((ext_vector_type(2))) float v2f;
typedef __attribute__((ext_vector_type(8))) float v8f;

#define N_SPIN 64
#define D_BOND 128
#define BATCH  1024

// ---- LDS layout (in floats) ----
// M buffers: COLUMN-major (col c holds M[0..127][c] contiguously along K),
// padded column stride 132 dwords -> B-operand ds_load_b64 is a natural pair
// and the wave's 32 lanes hit 64 distinct banks.
#define MT_STRIDE  132
#define M_FLOATS   (128 * MT_STRIDE)         // 16896 floats = 66KB
// carry stripes: per wave 16 rows row-major, padded stride 132 dwords.
#define C_STRIDE   132
#define CARRY_OFF  (2 * M_FLOATS)
#define CARRY_WAVE (16 * C_STRIDE)           // 2112 floats per wave
#define TRACE_OFF  (CARRY_OFF + 8 * CARRY_WAVE)
#define SMEM_FLOATS (TRACE_OFF + 4)          // ~203KB < 320KB/WGP

// ---------------------------------------------------------------------------
// One-time: transpose stack[i, k, s, c] -> wsT[(i*2+s)][c][k]  (col-major M)
// so the per-step DMA can copy contiguous K-runs straight into LDS columns.
// ---------------------------------------------------------------------------
__global__ __launch_bounds__(256)
void transpose_stack_kernel(const float* __restrict__ stack,
                            float* __restrict__ wsT) {
  const int m    = blockIdx.x;          // 0..127 : (i,s)
  const int i    = m >> 1;
  const int s    = m & 1;
  const int lane = threadIdx.x & 31;
  const int wave = threadIdx.x >> 5;
  const float* in  = stack + (size_t)(i * 256 + s) * 128;  // + k*256 + c
  float*       out = wsT   + (size_t)m * (128 * 128);      // + c*128 + k
#pragma unroll
  for (int it = 0; it < 16; ++it) {
    const int c = wave * 16 + it;       // column 0..127
    const int k = lane * 4;
    float4 v;
    v.x = in[(size_t)(k + 0) * 256 + c];
    v.y = in[(size_t)(k + 1) * 256 + c];
    v.z = in[(size_t)(k + 2) * 256 + c];
    v.w = in[(size_t)(k + 3) * 256 + c];
    *(float4*)(out + (size_t)c * 128 + k) = v;             // coalesced along k
  }
}

// Async-prefetch one transposed 128x128 matrix (64KB) into an LDS buffer.
// 8 waves x 16 iters x 32 lanes x 16B = 64KB; 16 ASYNCcnt ops per wave.
__device__ __forceinline__ void prefetch_Mt_async(const float* __restrict__ wsT,
                                                  int i, int s,
                                                  uint32_t lds_base_bytes,
                                                  int wave, int lane) {
  const float* base = wsT + (size_t)(i * 2 + s) * (128 * 128);
#pragma unroll
  for (int it = 0; it < 16; ++it) {
    int t   = wave * 512 + it * 32 + lane;   // 16B chunk id, 0..4095
    int c   = t >> 5;                        // column 0..127
    int off = t & 31;                        // 16B unit along K
    uint64_t ga = (uint64_t)(uintptr_t)(base + (size_t)c * 128 + off * 4);
    uint32_t la = lds_base_bytes + (uint32_t)(c * (MT_STRIDE * 4) + off * 16);
    asm volatile("global_load_async_to_lds_b128 %0, %1, off"
                 :: "v"(la), "v"(ga)
                 : "memory");
  }
}

__global__ __launch_bounds__(256)
void mps_chain_kernel(const float* __restrict__ x,
                      const float* __restrict__ stack,
                      const float* __restrict__ wsT,
                      float* __restrict__ out) {
  __shared__ __align__(16) float smem[SMEM_FLOATS];

  const int b     = blockIdx.x;
  const int lane  = threadIdx.x & 31;
  const int wave  = threadIdx.x >> 5;
  const int lrow  = lane & 15;   // M (A) / N (B,C) index within tile
  const int lhalf = lane >> 4;   // which half of the wave

  float* carry = smem + CARRY_OFF + wave * CARRY_WAVE;
  const uint32_t smem_base = (uint32_t)(uintptr_t)(void*)smem;  // LDS byte offset

  // ---- initial carry = M_0 stripe (rows 16w..16w+15), synchronous copy ----
  {
    const int s0 = x[b * N_SPIN + 0] > 0.0f ? 1 : 0;
    const float* base = stack + (size_t)s0 * 128;
#pragma unroll
    for (int it = 0; it < 16; ++it) {
      const int grow = wave * 16 + it;
      const float4 v = *(const float4*)(base + (size_t)grow * 256 + lane * 4);
      *(float4*)(carry + it * C_STRIDE + lane * 4) = v;
    }
  }

  // ---- prefetch M_1 (transposed) into buffer 0 ----
  {
    const int s1 = x[b * N_SPIN + 1] > 0.0f ? 1 : 0;
    prefetch_Mt_async(wsT, 1, s1, smem_base, wave, lane);
  }

  for (int i = 1; i < N_SPIN; ++i) {
    __syncthreads();  // all waves done reading buf[i&1] (used at step i-1)

    if (i < N_SPIN - 1) {
      const int sn = x[b * N_SPIN + i + 1] > 0.0f ? 1 : 0;
      prefetch_Mt_async(wsT, i + 1, sn,
                        smem_base + (uint32_t)((i & 1) * M_FLOATS * 4),
                        wave, lane);
      // 16 new ops outstanding; <=16 guarantees M_i's 16 ops completed
      asm volatile("s_wait_asynccnt 16" ::: "memory");
    } else {
      asm volatile("s_wait_asynccnt 0" ::: "memory");
    }
    __syncthreads();  // every wave's share of M_i is now visible in LDS

    const float* Mb = smem + ((i + 1) & 1) * M_FLOATS;  // col-major M_i

    v8f acc[8];
    const v8f zero = {0.f, 0.f, 0.f, 0.f, 0.f, 0.f, 0.f, 0.f};
#pragma unroll
    for (int n = 0; n < 8; ++n) acc[n] = zero;

    // new_stripe(16x128) = old_stripe(16x128) @ M(128x128)
    // K in chunks of 4 via V_WMMA_F32_16X16X4_F32
#pragma unroll 8
    for (int k = 0; k < 32; ++k) {
      const int kr = 4 * k + 2 * lhalf;  // this lane's K pair base
      // A pair: (old[r][kr], old[r][kr+1]) -- contiguous in row-major carry
      v2f a = *(const v2f*)(carry + lrow * C_STRIDE + kr);
      const float* bp = Mb + lrow * MT_STRIDE + kr;  // + n*16*MT_STRIDE
#pragma unroll
      for (int n = 0; n < 8; ++n) {
        // B pair: (M[kr][16n+c], M[kr+1][16n+c]) -- contiguous in col-major M
        v2f bv = *(const v2f*)(bp + n * 16 * MT_STRIDE);
        acc[n] = __builtin_amdgcn_wmma_f32_16x16x4_f32(
            /*neg_a=*/false, a, /*neg_b=*/false, bv,
            /*c_mod=*/(short)0, acc[n],
            /*reuse_a=*/false, /*reuse_b=*/false);
      }
    }

    // write back new carry: C/D layout -> row-major stripe (in-order per wave)
#pragma unroll
    for (int n = 0; n < 8; ++n) {
#pragma unroll
      for (int j = 0; j < 8; ++j) {
        carry[(j + 8 * lhalf) * C_STRIDE + 16 * n + lrow] = acc[n][j];
      }
    }
  }

  // ---- trace ----
  float* tr = smem + TRACE_OFF;
  if (threadIdx.x == 0) *tr = 0.0f;
  __syncthreads();
  if (lane < 16) {
    // wave w holds global rows 16w..16w+15; diagonal col = 16w + r
    float v = carry[lane * C_STRIDE + wave * 16 + lane];
    atomicAdd(tr, v);
  }
  __syncthreads();
  if (threadIdx.x == 0) out[b] = *tr;
}

extern "C" void kernel_launch(void* const* d_in, const int* in_sizes, int n_in,
                              void* d_out, int out_size, void* d_ws, size_t ws_size,
                              hipStream_t stream) {
  const float* x     = (const float*)d_in[0];   // (1024, 64) +-1.0
  const float* stack = (const float*)d_in[1];   // (64, 128, 2, 128)
  float* wsT = (float*)d_ws;                    // 64*2*128*128 floats = 8.4MB
  float* out = (float*)d_out;                   // (1024,)
  (void)in_sizes; (void)n_in; (void)out_size; (void)ws_size;

  // One-time (per launch) transpose of the weight stack into scratch.
  transpose_stack_kernel<<<dim3(128), dim3(256), 0, stream>>>(stack, wsT);
  // Main MPS chain contraction.
  mps_chain_kernel<<<dim3(BATCH), dim3(256), 0, stream>>>(x, stack, wsT, out);
}